// StepwiseVectorizedConstantVelocityModel_55267639164999
// MI455X (gfx1250) — compile-verified
//
#include <hip/hip_runtime.h>
#include <math.h>

#define N_POINTS 128
#define STEPS    50
#define TILE     16
#define NTILES   (N_POINTS / TILE)             // 8
#define NPAIRS   (NTILES * (NTILES + 1) / 2)   // 36 (ti <= tj)

typedef __attribute__((ext_vector_type(2))) float v2f;
typedef __attribute__((ext_vector_type(8))) float v8f;

// ---------------- workspace layout (bytes) ----------------
// [0,      512)  : double evt_part[64]        (per-block event partial sums)
// [512,  15360)  : double pair_part[1800]     (per (step,tilepair) partial sums)
// [15360,15616)  : int    firstidx[64]
// [16384, ... )  : float  Zs[128*2*51]        (trajectory prefix sums)

// K1: build Z_steps exactly like the reference:
//   Zs[i,d,0] = z0;  Zs[i,d,k>=1] = cumsum_{s<k}(z0 + v0[s]*dt)   (sequential fp32)
__global__ void k_pre(const float* __restrict__ z0, const float* __restrict__ v0,
                      const float* __restrict__ tn,
                      float* __restrict__ Zs, int* __restrict__ firstidx) {
  int tid = threadIdx.x;
  float dt = tn[0] / (float)STEPS;
  if (tid < N_POINTS) {
    for (int d = 0; d < 2; ++d) {
      float z0v = z0[tid * 2 + d];
      float* Zrow = Zs + (tid * 2 + d) * (STEPS + 1);
      const float* vrow = v0 + (tid * 2 + d) * STEPS;
      Zrow[0] = z0v;
      float acc = 0.0f;
      for (int k = 1; k <= STEPS; ++k) {
        acc += z0v + vrow[k - 1] * dt;   // cumsum(z0 + v0*dt), same order as jnp
        Zrow[k] = acc;
      }
    }
  }
  if (tid < STEPS) firstidx[tid] = 0x7FFFFFFF;
}

// K2: per-event intensity + segment-min of first event index per step.
__global__ void k_events(const float* __restrict__ data, int M,
                         const float* __restrict__ beta, const float* __restrict__ tn,
                         const float* __restrict__ Zs, const float* __restrict__ v0,
                         int* __restrict__ firstidx, double* __restrict__ evt_part) {
  __shared__ double red[256];
  int m = blockIdx.x * blockDim.x + threadIdx.x;
  double val = 0.0;
  if (m < M) {
    float dt = tn[0] / (float)STEPS;
    float b0 = beta[0];
    float srcf = data[3 * m + 0];
    float dstf = data[3 * m + 1];
    float t    = data[3 * m + 2];
    int ii = (int)floorf(srcf);
    int jj = (int)floorf(dstf);
    float denom = dt + dt * 0.001f;
    float stepf = floorf(t / denom);
    int k = (int)stepf;
    float delta = t - stepf * dt;
    float zi0 = Zs[(ii * 2 + 0) * (STEPS + 1) + k] + v0[(ii * 2 + 0) * STEPS + k] * delta;
    float zi1 = Zs[(ii * 2 + 1) * (STEPS + 1) + k] + v0[(ii * 2 + 1) * STEPS + k] * delta;
    float zj0 = Zs[(jj * 2 + 0) * (STEPS + 1) + k] + v0[(jj * 2 + 0) * STEPS + k] * delta;
    float zj1 = Zs[(jj * 2 + 1) * (STEPS + 1) + k] + v0[(jj * 2 + 1) * STEPS + k] * delta;
    float dx = zi0 - zj0, dy = zi1 - zj1;
    val = (double)(b0 - (dx * dx + dy * dy));
    atomicMin(&firstidx[k], m);     // order-invariant -> deterministic
  }
  red[threadIdx.x] = val;
  __syncthreads();
  for (int off = blockDim.x >> 1; off > 0; off >>= 1) {
    if (threadIdx.x < (unsigned)off) red[threadIdx.x] += red[threadIdx.x + off];
    __syncthreads();
  }
  if (threadIdx.x == 0) evt_part[blockIdx.x] = red[0];
}

// K3: one wave (32 lanes) per (step, 16x16 tile-pair). Gram matrices via
// V_WMMA_F32_16X16X4_F32 (K=2 zero-padded to 4), then per-pair erf integral.
__global__ void k_pairs(const float* __restrict__ data, int M,
                        const float* __restrict__ beta, const float* __restrict__ tn,
                        const float* __restrict__ Zs, const float* __restrict__ v0,
                        const int* __restrict__ firstidx, double* __restrict__ pair_part) {
  int s = blockIdx.x;                 // step
  int p = blockIdx.y;                 // tile-pair index
  int lane = threadIdx.x;
  __shared__ float ppA[TILE], vvA[TILE], pvA[TILE], ppB[TILE], vvB[TILE], pvB[TILE];
  __shared__ double red[32];

  int fi = firstidx[s];
  int outIdx = s * NPAIRS + p;
  if (fi >= M) {                      // valid[s] == false: segment contributes 0
    if (lane == 0) pair_part[outIdx] = 0.0;
    return;
  }
  // pair index -> (ti, tj) with ti <= tj
  int ti = 0, q = p;
  while (q >= NTILES - ti) { q -= NTILES - ti; ++ti; }
  int tj = ti + q;

  float b0 = beta[0];
  float ts = data[3 * fi + 2];
  float tf;
  if (s == STEPS - 1) {
    tf = data[3 * (M - 1) + 2];
  } else {
    int f2 = firstidx[s + 1];
    if (f2 > M - 1) f2 = M - 1;       // safe = min(first_idx, m-1)
    tf = data[3 * f2 + 2];
  }

  // A rows = tile ti, B cols = tile tj.  Lanes 0-15 carry K=0,1; lanes 16-31 are
  // the K=2,3 half which we zero-pad.
  v2f aP = {0.f, 0.f}, aV = {0.f, 0.f}, bP = {0.f, 0.f}, bV = {0.f, 0.f};
  if (lane < TILE) {
    int ri = ti * TILE + lane;
    int rj = tj * TILE + lane;
    aP.x = Zs[(ri * 2 + 0) * (STEPS + 1) + s];
    aP.y = Zs[(ri * 2 + 1) * (STEPS + 1) + s];
    aV.x = v0[(ri * 2 + 0) * STEPS + s];
    aV.y = v0[(ri * 2 + 1) * STEPS + s];
    bP.x = Zs[(rj * 2 + 0) * (STEPS + 1) + s];
    bP.y = Zs[(rj * 2 + 1) * (STEPS + 1) + s];
    bV.x = v0[(rj * 2 + 0) * STEPS + s];
    bV.y = v0[(rj * 2 + 1) * STEPS + s];
    ppA[lane] = aP.x * aP.x + aP.y * aP.y;
    vvA[lane] = aV.x * aV.x + aV.y * aV.y;
    pvA[lane] = aP.x * aV.x + aP.y * aV.y;
    ppB[lane] = bP.x * bP.x + bP.y * bP.y;
    vvB[lane] = bV.x * bV.x + bV.y * bV.y;
    pvB[lane] = bP.x * bV.x + bP.y * bV.y;
  }
  __syncthreads();                    // re-converge: EXEC all-ones for WMMA

  v8f zero = {};
  // D = A x B (+0): fp32 Gram tiles P·P^T, V·V^T, P·V^T, V·P^T
  v8f gPP = __builtin_amdgcn_wmma_f32_16x16x4_f32(false, aP, false, bP, (short)0, zero, false, false);
  v8f gVV = __builtin_amdgcn_wmma_f32_16x16x4_f32(false, aV, false, bV, (short)0, zero, false, false);
  v8f gPV = __builtin_amdgcn_wmma_f32_16x16x4_f32(false, aP, false, bV, (short)0, zero, false, false);
  v8f gVP = __builtin_amdgcn_wmma_f32_16x16x4_f32(false, aV, false, bP, (short)0, zero, false, false);

  const float SQRTPI = 1.7724538509055160f;
  int n = lane & 15;                          // N column within tile (C/D layout)
  int mbase = (lane < TILE) ? 0 : 8;          // lanes>=16 hold rows M=8..15
  double lsum = 0.0;
  for (int r = 0; r < 8; ++r) {
    int Mrow = mbase + r;
    int ig = ti * TILE + Mrow;
    int jg = tj * TILE + n;
    if (ig < jg) {                            // triu(k=1) mask
      float a  = vvA[Mrow] + vvB[n] - 2.0f * gVV[r];
      a = fmaxf(a, 1e-10f);
      float c  = ppA[Mrow] + ppB[n] - 2.0f * gPP[r];
      float bb = 2.0f * (pvA[Mrow] + pvB[n] - gPV[r] - gVP[r]);
      float sa = sqrtf(a);
      float shift = bb / (2.0f * sa);
      float pref = SQRTPI / (2.0f * sa) * expf(b0 - c + shift * shift);
      float val = pref * (erff(sa * tf + shift) - erff(sa * ts + shift));
      lsum += (double)val;
    }
  }
  red[lane] = lsum;
  __syncthreads();
  if (lane == 0) {
    double t2 = 0.0;
    for (int l = 0; l < 32; ++l) t2 += red[l];
    pair_part[outIdx] = t2;
  }
}

// K4: deterministic fixed-order final reduction -> scalar output.
__global__ void k_final(const double* __restrict__ evt_part, int nEvtBlocks,
                        const double* __restrict__ pair_part, int nPair,
                        float* __restrict__ out) {
  if (threadIdx.x == 0 && blockIdx.x == 0) {
    double ev = 0.0;
    for (int i = 0; i < nEvtBlocks; ++i) ev += evt_part[i];
    double ne = 0.0;
    for (int i = 0; i < nPair; ++i) ne += pair_part[i];
    out[0] = (float)(ev - ne);
  }
}

extern "C" void kernel_launch(void* const* d_in, const int* in_sizes, int n_in,
                              void* d_out, int out_size, void* d_ws, size_t ws_size,
                              hipStream_t stream) {
  const float* data = (const float*)d_in[0];
  const float* tn   = (const float*)d_in[2];
  const float* beta = (const float*)d_in[3];
  const float* z0   = (const float*)d_in[4];
  const float* v0   = (const float*)d_in[5];
  int M = in_sizes[0] / 3;

  char* ws = (char*)d_ws;
  double* evt_part  = (double*)(ws + 0);
  double* pair_part = (double*)(ws + 512);
  int*    firstidx  = (int*)   (ws + 15360);
  float*  Zs        = (float*) (ws + 16384);

  k_pre<<<1, 128, 0, stream>>>(z0, v0, tn, Zs, firstidx);

  int nEvtBlocks = (M + 255) / 256;
  k_events<<<nEvtBlocks, 256, 0, stream>>>(data, M, beta, tn, Zs, v0, firstidx, evt_part);

  dim3 grid(STEPS, NPAIRS);
  k_pairs<<<grid, 32, 0, stream>>>(data, M, beta, tn, Zs, v0, firstidx, pair_part);

  k_final<<<1, 32, 0, stream>>>(evt_part, nEvtBlocks, pair_part, STEPS * NPAIRS, (float*)d_out);
}